// minGRU2_20237885899461
// MI455X (gfx1250) — compile-verified
//
#include <hip/hip_runtime.h>

typedef __attribute__((ext_vector_type(2))) float v2f;
typedef __attribute__((ext_vector_type(8))) float v8f;

#define DIMI 128   // input channels
#define ODIM 256   // conv output channels
#define SEQL 8192  // sequence length
#define NB   8     // batch
#define CH   64    // channels per direction half

// ---------------- K1: EDM2 magnitude-preserving weight norm ----------------
// w_eff[o][i] = w[o][i] / (||w[o]|| + 1e-4*sqrt(128))
__global__ void wnorm_kernel(const float* __restrict__ w, float* __restrict__ we) {
    int o = blockIdx.x * blockDim.x + threadIdx.x;
    if (o >= ODIM) return;
    float s = 0.f;
#pragma unroll 8
    for (int i = 0; i < DIMI; ++i) { float v = w[o * DIMI + i]; s += v * v; }
    float inv = 1.f / (sqrtf(s) + 1.13137084989847604e-3f); // 1e-4*sqrt(128)
#pragma unroll 8
    for (int i = 0; i < DIMI; ++i) we[o * DIMI + i] = w[o * DIMI + i] * inv;
}

// ---------------- K2: WMMA GEMM + minGRU coefficient transform -------------
// Per block: one (batch, 16-wide L tile); 8 waves cover all 256 out channels
// as 4 fore + 4 back (h,g) tile pairs. Each wave accumulates two 16x16 f32
// tiles via V_WMMA_F32_16X16X4_F32 over K=128, then emits (a, b) coeffs:
//   z = sigmoid(g); s = sqrt(sech(g)+1); a = (1-z)*s; b = z*s*h
// using hardware v_exp/v_rcp/v_sqrt transcendentals.
__global__ void __launch_bounds__(256)
gemm_coef_kernel(const float* __restrict__ x, const float* __restrict__ we,
                 float2* __restrict__ coef) {
    const int lane = threadIdx.x & 31;
    const int wv   = threadIdx.x >> 5;   // 0..7
    const int dir  = wv >> 2;            // 0 = fore, 1 = back
    const int mg   = wv & 3;             // 0..3
    const int hb   = dir * 128 + mg * 16;    // h-channel tile base
    const int gb   = hb + 64;                // paired g-channel tile base
    const int n0   = blockIdx.x * 16;        // L tile
    const int bb   = blockIdx.y;             // batch

    const int ln    = lane & 15;
    const int khalf = (lane >> 4) << 1;      // lanes 16-31 handle K+2,K+3

    const float* xb  = x + (size_t)bb * DIMI * SEQL;
    const float* wrh = we + (size_t)(hb + ln) * DIMI;
    const float* wrg = we + (size_t)(gb + ln) * DIMI;

    v8f ch = {};   // h accumulator (16x16 f32)
    v8f cg = {};   // g accumulator

#pragma unroll 4
    for (int k = 0; k < DIMI; k += 4) {
        // A fragments: 16x4 f32, contiguous K pair per lane -> 8B loads
        const float2 th = *(const float2*)(wrh + k + khalf);
        const float2 tg = *(const float2*)(wrg + k + khalf);
        v2f ah = { th.x, th.y };
        v2f ag = { tg.x, tg.y };
        // B fragment: 4x16 f32 slice of x (rows K+khalf, K+khalf+1)
        const float b0 = xb[(size_t)(k + khalf) * SEQL + n0 + ln];
        const float b1 = xb[(size_t)(k + khalf + 1) * SEQL + n0 + ln];
        v2f bf = { b0, b1 };
        ch = __builtin_amdgcn_wmma_f32_16x16x4_f32(false, ah, false, bf,
                                                   (short)0, ch, false, false);
        cg = __builtin_amdgcn_wmma_f32_16x16x4_f32(false, ag, false, bf,
                                                   (short)0, cg, false, false);
    }

    const int mrow = (lane >> 4) << 3;   // C layout: lanes>=16 hold M=r+8
    const int n    = n0 + ln;
#pragma unroll
    for (int r = 0; r < 8; ++r) {
        const int c = (hb & 63) + mrow + r;     // coefficient channel 0..63
        const float h = ch[r];
        const float g = cg[r];
        const float eg  = __expf(g);
        const float emg = __expf(-g);
        // Robust at eg/emg -> inf: rcp(inf)=0 gives z/oz/s -> {1,0,1}.
        const float z  = __builtin_amdgcn_rcpf(1.f + emg);   // sigmoid(g)
        const float oz = __builtin_amdgcn_rcpf(1.f + eg);    // 1 - z
        const float s  = __builtin_amdgcn_sqrtf(
            2.f * __builtin_amdgcn_rcpf(eg + emg) + 1.f);    // sqrt(sech(g)+1)
        coef[((size_t)(dir * NB + bb) * CH + c) * SEQL + n] =
            make_float2(oz * s, z * s * h);
    }
}

// ---------------- K3: affine parallel scan along L -------------------------
// One block per (dir, batch, channel). 256 threads x 32 elements in regs.
// h_t = a_t * h_{t-1} + b_t, h_{-1} = 0. DIR=1 scans reversed L and writes
// back un-flipped; output layout = mp_cat (weights exactly 1.0).
template <int DIR>
__global__ void __launch_bounds__(256)
scan_kernel(const float2* __restrict__ coef, float* __restrict__ out) {
    const int tid = threadIdx.x;
    const int c   = blockIdx.x;
    const int bb  = blockIdx.y;
    const size_t base  = ((size_t)(DIR * NB + bb) * CH + c) * SEQL;
    const int    lbase = DIR ? (SEQL - 32 - tid * 32) : (tid * 32);

    float2 e[32];
#pragma unroll
    for (int j = 0; j < 32; ++j) e[j] = coef[base + lbase + j];

    // local affine composition over this thread's chunk (in scan order)
    float A = 1.f, Bv = 0.f;
#pragma unroll
    for (int s = 0; s < 32; ++s) {
        const int j = DIR ? (31 - s) : s;
        A  = e[j].x * A;
        Bv = e[j].x * Bv + e[j].y;
    }

    __shared__ float sA[256];
    __shared__ float sB[256];
    sA[tid] = A; sB[tid] = Bv;
    __syncthreads();
    for (int off = 1; off < 256; off <<= 1) {
        float pa = 1.f, pb = 0.f;
        if (tid >= off) { pa = sA[tid - off]; pb = sB[tid - off]; }
        __syncthreads();
        if (tid >= off) {
            const float ca = sA[tid], cb = sB[tid];
            sA[tid] = ca * pa;
            sB[tid] = ca * pb + cb;
        }
        __syncthreads();
    }
    // carry-in: inclusive scan of previous thread applied to h=0 -> B prefix
    float h = (tid == 0) ? 0.f : sB[tid - 1];

#pragma unroll
    for (int s = 0; s < 32; ++s) {
        const int j = DIR ? (31 - s) : s;
        h = e[j].x * h + e[j].y;
        e[j].x = h;
    }

    float* op = out + ((size_t)bb * 128 + DIR * 64 + c) * SEQL + lbase;
#pragma unroll
    for (int j = 0; j < 32; ++j) op[j] = e[j].x;
}

// ---------------- launcher -------------------------------------------------
extern "C" void kernel_launch(void* const* d_in, const int* in_sizes, int n_in,
                              void* d_out, int out_size, void* d_ws, size_t ws_size,
                              hipStream_t stream) {
    const float* x = (const float*)d_in[0];  // [8,128,8192]
    const float* w = (const float*)d_in[1];  // [256,128,1]
    float* out = (float*)d_out;              // [8,128,8192]

    float*  we   = (float*)d_ws;                                   // 128 KB
    float2* coef = (float2*)((char*)d_ws + ODIM * DIMI * sizeof(float)); // 64 MB

    wnorm_kernel<<<1, 256, 0, stream>>>(w, we);
    gemm_coef_kernel<<<dim3(SEQL / 16, NB), 256, 0, stream>>>(x, we, coef);
    scan_kernel<0><<<dim3(CH, NB), 256, 0, stream>>>(coef, out);
    scan_kernel<1><<<dim3(CH, NB), 256, 0, stream>>>(coef, out);
}